// MultiplicativeAttention_56478819942409
// MI455X (gfx1250) — compile-verified
//
#include <hip/hip_runtime.h>
#include <hip/hip_bf16.h>

typedef float v2f __attribute__((ext_vector_type(2)));
typedef float v8f __attribute__((ext_vector_type(8)));

#define BB 32
#define TT 4096
#define DD 1024

// ---------------------------------------------------------------------------
// Kernel 1: mid[b][k] = sum_q W[k][q] * query[b][q]
// One wave (32 threads) per 16x16 (k,b) tile, v_wmma_f32_16x16x4_f32 over q.
// A (16x4 f32): lanes 0-15 hold M=0..15 with VGPR0=K0,VGPR1=K1;
//               lanes 16-31 hold M=0..15 with VGPR0=K2,VGPR1=K3.
// B (4x16 f32): lanes 0-15 N=0..15 K0/K1; lanes 16-31 N=0..15 K2/K3.
// D (16x16 f32, 8 VGPRs): VGPR j -> M=j (lanes 0-15), M=j+8 (lanes 16-31).
// ---------------------------------------------------------------------------
__global__ void mid_gemm_wmma(const float* __restrict__ W,
                              const float* __restrict__ query,
                              float* __restrict__ mid) {
  const int kt   = blockIdx.x * 16;   // k tile (M)
  const int bt   = blockIdx.y * 16;   // b tile (N)
  const int lane = threadIdx.x;       // wave32
  const int half = lane >> 4;         // 0: K+0/K+1, 1: K+2/K+3
  const int l16  = lane & 15;

  const float* __restrict__ Wrow = W + (size_t)(kt + l16) * DD;     // A row
  const float* __restrict__ qrow = query + (size_t)(bt + l16) * DD; // B col

  v8f acc = {};
  for (int q = 0; q < DD; q += 4) {
    const int qa = q + half * 2;
    v2f a, b;
    a[0] = Wrow[qa];  a[1] = Wrow[qa + 1];
    b[0] = qrow[qa];  b[1] = qrow[qa + 1];
    acc = __builtin_amdgcn_wmma_f32_16x16x4_f32(
        /*neg_a=*/false, a, /*neg_b=*/false, b,
        /*c_mod=*/(short)0, acc, /*reuse_a=*/false, /*reuse_b=*/false);
  }

  // Store: mid is laid out [B][D] so kernel 2 reads it contiguously.
  const int b_idx = bt + l16;          // N = lane (0-15) / lane-16 (16-31)
  const int kbase = kt + half * 8;     // M = j (+8 for upper half-lanes)
#pragma unroll
  for (int j = 0; j < 8; ++j) {
    mid[(size_t)b_idx * DD + kbase + j] = acc[j];
  }
}

// ---------------------------------------------------------------------------
// Kernel 2: scores[b][t] = dot(key[b][t][:], mid[b][:]) with mask -> -1e18
// One wave per row; float4 streaming loads of key (512 B / instr / wave).
// ---------------------------------------------------------------------------
__global__ void scores_kernel(const float* __restrict__ key,
                              const int* __restrict__ mask,
                              const float* __restrict__ mid,
                              float* __restrict__ scores) {
  const int ROWS = 64;                          // rows per block
  const int b  = blockIdx.x / (TT / ROWS);
  const int t0 = (blockIdx.x % (TT / ROWS)) * ROWS;

  __shared__ float smid[DD];
  for (int i = threadIdx.x; i < DD; i += 256) smid[i] = mid[(size_t)b * DD + i];
  __syncthreads();

  const int wave = threadIdx.x >> 5;  // 8 waves
  const int lane = threadIdx.x & 31;
  const float4* __restrict__ smid4 = (const float4*)smid;

  for (int r = wave; r < ROWS; r += 8) {
    const int t = t0 + r;
    const float4* __restrict__ krow =
        (const float4*)(key + ((size_t)b * TT + t) * DD);
    // prefetch next row for this wave (speculative, gfx1250 global_prefetch_b8)
    if (r + 8 < ROWS) {
      __builtin_prefetch(key + ((size_t)b * TT + t + 8) * DD + lane * 16, 0, 0);
    }
    float acc = 0.f;
#pragma unroll
    for (int i = 0; i < 8; ++i) {
      const float4 kv = krow[i * 32 + lane];
      const float4 mv = smid4[i * 32 + lane];
      acc += kv.x * mv.x + kv.y * mv.y + kv.z * mv.z + kv.w * mv.w;
    }
#pragma unroll
    for (int off = 16; off > 0; off >>= 1) acc += __shfl_xor(acc, off, 32);
    if (lane == 0) {
      scores[(size_t)b * TT + t] = (mask[(size_t)b * TT + t] == 0) ? -1e18f : acc;
    }
  }
}

// ---------------------------------------------------------------------------
// Kernel 3: per-batch softmax over T, in place on scores.
// ---------------------------------------------------------------------------
__global__ void softmax_kernel(float* __restrict__ scores) {
  const int b = blockIdx.x;
  __shared__ float red[256];
  float* __restrict__ row = scores + (size_t)b * TT;

  float m = -__builtin_inff();
  for (int t = threadIdx.x; t < TT; t += 256) m = fmaxf(m, row[t]);
  red[threadIdx.x] = m;
  __syncthreads();
  for (int s = 128; s > 0; s >>= 1) {
    if (threadIdx.x < s) red[threadIdx.x] = fmaxf(red[threadIdx.x], red[threadIdx.x + s]);
    __syncthreads();
  }
  m = red[0];
  __syncthreads();

  float l = 0.f;
  for (int t = threadIdx.x; t < TT; t += 256) l += __expf(row[t] - m);
  red[threadIdx.x] = l;
  __syncthreads();
  for (int s = 128; s > 0; s >>= 1) {
    if (threadIdx.x < s) red[threadIdx.x] += red[threadIdx.x + s];
    __syncthreads();
  }
  const float inv = 1.0f / red[0];

  for (int t = threadIdx.x; t < TT; t += 256) row[t] = __expf(row[t] - m) * inv;
}

// ---------------------------------------------------------------------------
// Kernel 4: partial[b][c][v] = sum_{t in chunk c} prob[b][t] * value[b][t][v]
// 32 chunks x 32 batches = 1024 blocks; each thread owns 4 v-columns.
// Per t: the block reads one contiguous 4 KB value row (fully coalesced).
// ---------------------------------------------------------------------------
__global__ void pv_partial_kernel(const float* __restrict__ value,
                                  const float* __restrict__ prob,
                                  float* __restrict__ part) {
  const int TC = 32, CH = TT / TC;   // 128 t per chunk
  const int b  = blockIdx.x / TC;
  const int c  = blockIdx.x % TC;
  const int t0 = c * CH;

  __shared__ float sp[CH];
  for (int i = threadIdx.x; i < CH; i += 256) sp[i] = prob[(size_t)b * TT + t0 + i];
  __syncthreads();

  const int v0 = threadIdx.x * 4;
  float4 acc = {0.f, 0.f, 0.f, 0.f};
  for (int tt = 0; tt < CH; ++tt) {
    const float p = sp[tt];
    const float4 v =
        *(const float4*)(value + ((size_t)b * TT + t0 + tt) * DD + v0);
    acc.x += p * v.x;
    acc.y += p * v.y;
    acc.z += p * v.z;
    acc.w += p * v.w;
  }
  *(float4*)(part + (size_t)blockIdx.x * DD + v0) = acc;
}

// ---------------------------------------------------------------------------
// Kernel 5: out[b][v] = sum_c partial[b][c][v]  (deterministic reduction)
// ---------------------------------------------------------------------------
__global__ void reduce_kernel(const float* __restrict__ part,
                              float* __restrict__ out) {
  const int b  = blockIdx.x;
  const int v0 = threadIdx.x * 4;
  float4 acc = {0.f, 0.f, 0.f, 0.f};
#pragma unroll 4
  for (int c = 0; c < 32; ++c) {
    const float4 p = *(const float4*)(part + (size_t)(b * 32 + c) * DD + v0);
    acc.x += p.x; acc.y += p.y; acc.z += p.z; acc.w += p.w;
  }
  *(float4*)(out + (size_t)b * DD + v0) = acc;
}

// ---------------------------------------------------------------------------
extern "C" void kernel_launch(void* const* d_in, const int* in_sizes, int n_in,
                              void* d_out, int out_size, void* d_ws, size_t ws_size,
                              hipStream_t stream) {
  const float* query = (const float*)d_in[0];   // [B, D]
  const float* key   = (const float*)d_in[1];   // [B, T, D]
  const float* value = (const float*)d_in[2];   // [B, T, D]
  const int*   mask  = (const int*)d_in[3];     // [B, T]
  const float* W     = (const float*)d_in[4];   // [D, D]
  float* out = (float*)d_out;                   // [B, D]

  float* ws     = (float*)d_ws;
  float* mid    = ws;                           // 32*1024          = 128 KB
  float* scores = mid + (size_t)BB * DD;        // 32*4096          = 512 KB
  float* part   = scores + (size_t)BB * TT;     // 1024*1024 floats = 4 MB

  // 1) mid = W @ q^T  (WMMA fp32)
  mid_gemm_wmma<<<dim3(DD / 16, BB / 16), 32, 0, stream>>>(W, query, mid);

  // 2) masked scores (streams key, 512 MB)
  scores_kernel<<<BB * (TT / 64), 256, 0, stream>>>(key, mask, mid, scores);

  // 3) softmax per batch
  softmax_kernel<<<BB, 256, 0, stream>>>(scores);

  // 4) prob @ value partials (streams value, 512 MB)
  pv_partial_kernel<<<BB * 32, 256, 0, stream>>>(value, scores, part);

  // 5) deterministic combine -> out
  reduce_kernel<<<BB, 256, 0, stream>>>(part, out);
}